// SpGAT_11776800326013
// MI455X (gfx1250) — compile-verified
//
#include <hip/hip_runtime.h>
#include <math.h>

// ---------------------------------------------------------------------------
// SpGAT forward for MI455X (gfx1250, wave32).
//   GEMMs: V_WMMA_F32_16X16X4_F32 (exact f32 semantics; GEMMs are a tiny
//          fraction of the work — the kernel is L2-atomic bound, so a
//          precision downgrade buys nothing).
//   B matrix staged into LDS once per block via the Tensor Data Mover
//          (tensor_load_to_lds + s_wait_tensorcnt), zero-padded by TDM OOB.
//   Edge phases: global_atomic_add_f32 scatter into L2-resident buffers
//          (~90MB working set fits the 192MB L2).
// ---------------------------------------------------------------------------

typedef __attribute__((ext_vector_type(2))) float v2f;
typedef __attribute__((ext_vector_type(8))) float v8f;
typedef __attribute__((ext_vector_type(4))) unsigned int u32x4;
typedef __attribute__((ext_vector_type(4))) int i32x4;
typedef __attribute__((ext_vector_type(8))) int i32x8;

#define NFEAT  128
#define NHID   8
#define NHEADS 8
#define NCLASS 40
#define HCAT   64          // NHEADS * NHID
#define ALPHA  0.2f
#define EPS    1e-16f

// ---- repack W_heads (8,128,8) -> Wcat (128,64) row-major --------------------
__global__ void repack_wcat(const float* __restrict__ Wh, float* __restrict__ Wcat) {
    int idx = blockIdx.x * blockDim.x + threadIdx.x;
    if (idx >= NFEAT * HCAT) return;
    int k = idx >> 6, col = idx & 63;
    Wcat[idx] = Wh[(col >> 3) * (NFEAT * NHID) + k * NHID + (col & 7)];
}

// ---- f32 WMMA GEMM: C(nrows x NCOLS, ld=LDC) = A(nrows x K) * B(K x NCOLS) --
// B is DMAed to LDS (K x NPAD, zero-padded cols) by the TDM once per block.
// One wave per 16-row stripe (grid-stride); K in steps of 4, fully unrolled.
// Fragment layouts (wave32):
//   A 16x4 f32: lane L -> row M=L&15; v0 holds K=2*(L>>4), v1 holds K+1.
//   B 4x16 f32: lane L -> col N=L&15; v0 holds K=2*(L>>4), v1 holds K+1.
//   C/D 16x16 f32 (8 VGPRs): VGPR r, lanes 0-15 -> M=r, lanes 16-31 -> M=8+r.
template <int K, int NCOLS, int LDC>
__global__ void gemm_wmma(const float* __restrict__ A, const float* __restrict__ B,
                          float* __restrict__ C, int nrows) {
    constexpr int NPAD = (NCOLS + 15) & ~15;
    __shared__ float Bs[K * NPAD];

    // --- TDM: DMA B (K x NCOLS row-major) -> LDS tile (K x NPAD).
    // tensor_dim0 = NCOLS, tile_dim0 = NPAD: reads past the tensor edge return
    // zero, which zero-pads columns NCOLS..NPAD-1 in LDS for us.
    if ((threadIdx.x >> 5) == 0) {   // one wave issues the DMA (EXEC ignored by TDM)
        unsigned long long gaddr = (unsigned long long)(const void*)B;
        u32x4 g0;
        g0[0] = 1u;                                          // count=1 (valid D#)
        g0[1] = 0u;                                          // lds_addr = 0 (Bs is sole LDS alloc)
        g0[2] = (unsigned)(gaddr & 0xFFFFFFFFu);             // global_addr[31:0]
        g0[3] = (unsigned)((gaddr >> 32) & 0x01FFFFFFu)      // global_addr[56:32]
                | (2u << 30);                                // type = 2 ("image")
        i32x8 g1;
        g1[0] = (int)(2u << 16);                             // data_size = 4 bytes
        g1[1] = (int)((unsigned)NCOLS << 16);                // tensor_dim0[15:0]
        g1[2] = (int)((unsigned)K << 16);                    // dim0 hi=0 | tensor_dim1[15:0]
        g1[3] = (int)((unsigned)NPAD << 16);                 // dim1 hi=0 | tile_dim0
        g1[4] = K;                                           // tile_dim1 | tile_dim2=0
        g1[5] = NCOLS;                                       // tensor_dim0_stride[31:0]
        g1[6] = 0;                                           // stride hi | dim1_stride lo
        g1[7] = 0;
        i32x4 gz = {0, 0, 0, 0};
#if defined(__clang_major__) && (__clang_major__ >= 23)
        i32x8 gz8 = {0, 0, 0, 0, 0, 0, 0, 0};
        __builtin_amdgcn_tensor_load_to_lds(g0, g1, gz, gz, gz8, 0);
#else
        __builtin_amdgcn_tensor_load_to_lds(g0, g1, gz, gz, 0);
#endif
        __builtin_amdgcn_s_wait_tensorcnt(0);
    }
    __syncthreads();

    const int lane  = threadIdx.x & 31;
    const int wave  = threadIdx.x >> 5;
    const int kh    = lane >> 4;          // K-half of the fragment (0 or 1)
    const int col16 = lane & 15;
    const int wavesPerBlock = blockDim.x >> 5;
    const int totalWaves    = gridDim.x * wavesPerBlock;
    const int rowTiles      = nrows >> 4;     // nrows is a multiple of 16

    for (int t = blockIdx.x * wavesPerBlock + wave; t < rowTiles; t += totalWaves) {
        const int row0 = t << 4;
        const float* aBase = A + (long)(row0 + col16) * K + 2 * kh;
        for (int n0 = 0; n0 < NCOLS; n0 += 16) {
            const int col = n0 + col16;
            const int bBase = 2 * kh * NPAD + col;
            v8f acc = (v8f){0.f, 0.f, 0.f, 0.f, 0.f, 0.f, 0.f, 0.f};
#pragma unroll
            for (int k = 0; k < K; k += 4) {
                v2f a;
                a.x = aBase[k];
                a.y = aBase[k + 1];
                v2f b;
                b.x = Bs[k * NPAD + bBase];
                b.y = Bs[k * NPAD + bBase + NPAD];
                acc = __builtin_amdgcn_wmma_f32_16x16x4_f32(
                          false, a, false, b, (short)0, acc, false, false);
            }
            if (col < NCOLS) {
#pragma unroll
                for (int r = 0; r < 8; ++r)
                    C[(long)(row0 + 8 * kh + r) * LDC + col] = acc[r];
            }
        }
    }
}

// ---- per-(node, head) attention score halves: s = h . a --------------------
__global__ void scores_heads(const float* __restrict__ h, const float* __restrict__ a,
                             float* __restrict__ ssrc, float* __restrict__ sdst, int n_nodes) {
    int gid = blockIdx.x * blockDim.x + threadIdx.x;
    if (gid >= n_nodes * NHEADS) return;
    int n = gid >> 3, hd = gid & 7;
    const float* hr = h + (long)n * HCAT + hd * NHID;
    const float* av = a + hd * (2 * NHID);
    float s0 = 0.f, s1 = 0.f;
#pragma unroll
    for (int c = 0; c < NHID; ++c) { s0 += hr[c] * av[c]; s1 += hr[c] * av[NHID + c]; }
    ssrc[gid] = s0;
    sdst[gid] = s1;
}

// ---- layer-1 edge scatter: one thread per (edge, head) ---------------------
__global__ void edge_pass1(const int* __restrict__ ei, const float* __restrict__ h,
                           const float* __restrict__ ssrc, const float* __restrict__ sdst,
                           float* __restrict__ rowsum, float* __restrict__ hp, int n_edges) {
    long gid = (long)blockIdx.x * blockDim.x + threadIdx.x;
    if (gid >= (long)n_edges * NHEADS) return;
    int e = (int)(gid >> 3), hd = (int)(gid & 7);
    int src = ei[e], dst = ei[n_edges + e];
    float score = ssrc[src * NHEADS + hd] + sdst[dst * NHEADS + hd];
    float lr = score >= 0.f ? score : ALPHA * score;
    float w  = expf(-lr);
    atomicAdd(&rowsum[src * NHEADS + hd], w);
    const float* hv  = h  + (long)dst * HCAT + hd * NHID;
    float*       out = hp + (long)src * HCAT + hd * NHID;
#pragma unroll
    for (int c = 0; c < NHID; ++c) atomicAdd(&out[c], w * hv[c]);
}

// ---- layer-1 normalize + ELU -> xc -----------------------------------------
__global__ void finalize1(const float* __restrict__ hp, const float* __restrict__ rowsum,
                          float* __restrict__ xc, int n_nodes) {
    long gid = (long)blockIdx.x * blockDim.x + threadIdx.x;
    if (gid >= (long)n_nodes * HCAT) return;
    int n = (int)(gid >> 6), f = (int)(gid & 63);
    float v = hp[gid] / (rowsum[n * NHEADS + (f >> 3)] + EPS);
    xc[gid] = v > 0.f ? v : expm1f(v);
}

// ---- layer-2 score halves ---------------------------------------------------
__global__ void scores_out(const float* __restrict__ h2, const float* __restrict__ a,
                           float* __restrict__ ssrc, float* __restrict__ sdst, int n_nodes) {
    int n = blockIdx.x * blockDim.x + threadIdx.x;
    if (n >= n_nodes) return;
    const float* hr = h2 + (long)n * NCLASS;
    float s0 = 0.f, s1 = 0.f;
#pragma unroll
    for (int c = 0; c < NCLASS; ++c) { s0 += hr[c] * a[c]; s1 += hr[c] * a[NCLASS + c]; }
    ssrc[n] = s0;
    sdst[n] = s1;
}

// ---- layer-2 edge scatter: 8 lanes per edge, 5 classes each ----------------
__global__ void edge_pass2(const int* __restrict__ ei, const float* __restrict__ h2,
                           const float* __restrict__ ssrc, const float* __restrict__ sdst,
                           float* __restrict__ rowsum, float* __restrict__ hp, int n_edges) {
    long gid = (long)blockIdx.x * blockDim.x + threadIdx.x;
    if (gid >= (long)n_edges * 8) return;
    int e = (int)(gid >> 3), t = (int)(gid & 7);
    int src = ei[e], dst = ei[n_edges + e];
    float score = ssrc[src] + sdst[dst];
    float lr = score >= 0.f ? score : ALPHA * score;
    float w  = expf(-lr);
    if (t == 0) atomicAdd(&rowsum[src], w);
    const float* hv  = h2 + (long)dst * NCLASS + t * 5;
    float*       out = hp + (long)src * NCLASS + t * 5;
#pragma unroll
    for (int c = 0; c < 5; ++c) atomicAdd(&out[c], w * hv[c]);
}

// ---- layer-2 normalize + ELU + log_softmax ---------------------------------
__global__ void finalize2(const float* __restrict__ hp, const float* __restrict__ rowsum,
                          float* __restrict__ out, int n_nodes) {
    int n = blockIdx.x * blockDim.x + threadIdx.x;
    if (n >= n_nodes) return;
    float rs = rowsum[n] + EPS;
    float v[NCLASS];
    float mx = -INFINITY;
#pragma unroll
    for (int c = 0; c < NCLASS; ++c) {
        float t = hp[(long)n * NCLASS + c] / rs;
        t = t > 0.f ? t : expm1f(t);
        v[c] = t;
        mx = fmaxf(mx, t);
    }
    float s = 0.f;
#pragma unroll
    for (int c = 0; c < NCLASS; ++c) s += expf(v[c] - mx);
    float lse = mx + logf(s);
#pragma unroll
    for (int c = 0; c < NCLASS; ++c) out[(long)n * NCLASS + c] = v[c] - lse;
}

// ---------------------------------------------------------------------------
extern "C" void kernel_launch(void* const* d_in, const int* in_sizes, int n_in,
                              void* d_out, int out_size, void* d_ws, size_t ws_size,
                              hipStream_t stream) {
    const float* x  = (const float*)d_in[0];
    const int*   ei = (const int*)d_in[1];
    const float* Wh = (const float*)d_in[2];
    const float* ah = (const float*)d_in[3];
    const float* Wo = (const float*)d_in[4];
    const float* ao = (const float*)d_in[5];
    float*       out = (float*)d_out;

    const int n_nodes = in_sizes[0] / NFEAT;   // 100000 (divisible by 16)
    const int n_edges = in_sizes[1] / 2;       // 1600000

    // workspace layout (floats); layer-2 buffers alias dead layer-1 buffers.
    float* ws = (float*)d_ws;
    size_t off = 0;
    float* wcat   = ws + off; off += (size_t)NFEAT * HCAT;
    float* h1     = ws + off; off += (size_t)n_nodes * HCAT;
    float* ssrc   = ws + off; off += (size_t)n_nodes * NHEADS;
    float* sdst   = ws + off; off += (size_t)n_nodes * NHEADS;
    float* rowsum = ws + off; off += (size_t)n_nodes * NHEADS;
    float* hp     = ws + off; off += (size_t)n_nodes * HCAT;
    float* xc     = ws + off; off += (size_t)n_nodes * HCAT;
    // aliases for layer 2 (h1/ssrc/sdst/rowsum/hp are dead by then)
    float* h2 = h1; float* ssrc2 = ssrc; float* sdst2 = sdst;
    float* rowsum2 = rowsum; float* hp2 = hp;

    const int TB = 256; // 8 waves of 32

    // zero layer-1 accumulators (workspace is not re-poisoned between replays)
    hipMemsetAsync(rowsum, 0, (size_t)n_nodes * NHEADS * sizeof(float), stream);
    hipMemsetAsync(hp,     0, (size_t)n_nodes * HCAT   * sizeof(float), stream);

    repack_wcat<<<(NFEAT * HCAT + TB - 1) / TB, TB, 0, stream>>>(Wh, wcat);

    const int rowTiles   = (n_nodes + 15) / 16;
    int gemmBlocks = (rowTiles + 7) / 8;       // 8 waves per block, grid-stride
    if (gemmBlocks > 512) gemmBlocks = 512;

    gemm_wmma<NFEAT, HCAT, HCAT><<<gemmBlocks, TB, 0, stream>>>(x, wcat, h1, n_nodes);
    scores_heads<<<(n_nodes * NHEADS + TB - 1) / TB, TB, 0, stream>>>(h1, ah, ssrc, sdst, n_nodes);
    edge_pass1<<<(int)(((long)n_edges * NHEADS + TB - 1) / TB), TB, 0, stream>>>(
        ei, h1, ssrc, sdst, rowsum, hp, n_edges);
    finalize1<<<(int)(((long)n_nodes * HCAT + TB - 1) / TB), TB, 0, stream>>>(
        hp, rowsum, xc, n_nodes);

    // zero layer-2 accumulators (they alias layer-1 data)
    hipMemsetAsync(rowsum2, 0, (size_t)n_nodes * sizeof(float), stream);
    hipMemsetAsync(hp2,     0, (size_t)n_nodes * NCLASS * sizeof(float), stream);

    gemm_wmma<HCAT, NCLASS, NCLASS><<<gemmBlocks, TB, 0, stream>>>(xc, Wo, h2, n_nodes);
    scores_out<<<(n_nodes + TB - 1) / TB, TB, 0, stream>>>(h2, ao, ssrc2, sdst2, n_nodes);
    edge_pass2<<<(int)(((long)n_edges * 8 + TB - 1) / TB), TB, 0, stream>>>(
        ei, h2, ssrc2, sdst2, rowsum2, hp2, n_edges);
    finalize2<<<(n_nodes + TB - 1) / TB, TB, 0, stream>>>(hp2, rowsum2, out, n_nodes);
}